// mLSTMLocalizationClassifier_34376918237589
// MI455X (gfx1250) — compile-verified
//
#include <hip/hip_runtime.h>

typedef __attribute__((ext_vector_type(2))) float v2f;
typedef __attribute__((ext_vector_type(8))) float v8f;

#define LMAX  32768
#define EMBD  32
#define HID   64
#define G4    256     // 4*HID
#define VOC   6
#define NTHR  320     // 10 waves: waves 0-7 -> gates, waves 8-9 -> hm + state

__device__ __forceinline__ float fast_sigmoid(float x) {
    return __builtin_amdgcn_rcpf(1.0f + __expf(-x));
}
__device__ __forceinline__ float fast_tanh(float x) {
    // tanh(x) = 1 - 2/(1+e^{2x}) ; __expf saturates cleanly at +-inf
    return 1.0f - 2.0f * __builtin_amdgcn_rcpf(1.0f + __expf(2.0f * x));
}

__global__ __launch_bounds__(NTHR)
void mlstm_seq_kernel(const int* __restrict__ seq,
                      const float* __restrict__ emb,
                      const float* __restrict__ w_im, const float* __restrict__ b_im,
                      const float* __restrict__ w_hm, const float* __restrict__ b_hm,
                      const float* __restrict__ W_ih, const float* __restrict__ b_ih,
                      const float* __restrict__ W_hh, const float* __restrict__ b_hh,
                      const float* __restrict__ W_out, const float* __restrict__ b_out,
                      float* __restrict__ out, int L)
{
    __shared__ float sh_embp[16 * EMBD];     // emb padded to 16 rows (A matrix)
    __shared__ float sh_xihb[VOC][G4];       // emb[v]@W_ih.T + b_ih + b_hh
    __shared__ float sh_xim[VOC][HID];       // emb[v]@w_im.T + b_im
    __shared__ float sh_h[HID];
    __shared__ float sh_mx[HID];
    __shared__ float sh_gates[G4];
    __shared__ float sh_c[HID];
    __shared__ unsigned char sh_seq[LMAX];

    const int tid  = (int)threadIdx.x;
    const int lane = tid & 31;
    const int wv   = tid >> 5;          // wave id 0..9
    const int l    = lane & 15;
    const int half = lane >> 4;

    // ---- stage 0: padded A matrix (emb rows 0..5, zeros 6..15) + init h ----
    for (int i = tid; i < 16 * EMBD; i += NTHR)
        sh_embp[i] = (i < VOC * EMBD) ? emb[i] : 0.0f;
    if (tid < HID) sh_h[tid] = 0.0f;
    __syncthreads();

    // ---- stage A: WMMA f32 16x16x4 precompute of the two vocab tables ----
    // jobs 0..15 : xihb column-tile n=job  (cols 16n..16n+15 of 256), B = W_ih.T
    // jobs 16..19: xim  column-tile n=job-16 (cols 16n..16n+15 of 64), B = w_im.T
    for (int jobi = 0; jobi < 2; ++jobi) {
        const int job = wv + jobi * 10;             // wave-uniform, 0..19
        const float* Bmat;
        int colBase;
        if (job < 16) { Bmat = W_ih; colBase = job * 16; }
        else          { Bmat = w_im; colBase = (job - 16) * 16; }
        const int col = colBase + l;

        v8f acc = {};
        #pragma unroll
        for (int k = 0; k < 8; ++k) {
            const int kb = 4 * k + 2 * half;        // ISA f32 A/B lane-half K split
            v2f a, b;
            a.x = sh_embp[l * EMBD + kb];
            a.y = sh_embp[l * EMBD + kb + 1];
            b.x = Bmat[col * EMBD + kb];            // B[k][col] = W[col][k]
            b.y = Bmat[col * EMBD + kb + 1];
            acc = __builtin_amdgcn_wmma_f32_16x16x4_f32(
                      false, a, false, b, (short)0, acc, false, false);
        }
        // D layout: VGPR j holds row M=j (lanes 0-15) / M=j+8 (lanes 16-31).
        // Only rows 0..5 (vocab) are valid -> lower lane half, VGPRs 0..5.
        if (half == 0) {
            if (job < 16) {
                const float bias = b_ih[col] + b_hh[col];
                sh_xihb[0][col] = acc[0] + bias;
                sh_xihb[1][col] = acc[1] + bias;
                sh_xihb[2][col] = acc[2] + bias;
                sh_xihb[3][col] = acc[3] + bias;
                sh_xihb[4][col] = acc[4] + bias;
                sh_xihb[5][col] = acc[5] + bias;
            } else {
                const float bias = b_im[col];
                sh_xim[0][col] = acc[0] + bias;
                sh_xim[1][col] = acc[1] + bias;
                sh_xim[2][col] = acc[2] + bias;
                sh_xim[3][col] = acc[3] + bias;
                sh_xim[4][col] = acc[4] + bias;
                sh_xim[5][col] = acc[5] + bias;
            }
        }
    }

    // ---- stage B: stage full token sequence in LDS (bytes, VOCAB<256) ----
    for (int i = tid; i < L; i += NTHR)
        sh_seq[i] = (unsigned char)seq[i];

    // ---- stage C: pin per-thread weight rows in VGPRs ----
    float wrow[HID];
    float bhm = 0.0f;
    if (tid < G4) {
        #pragma unroll
        for (int k = 0; k < HID; ++k) wrow[k] = W_hh[tid * HID + k];
    } else {
        const int j = tid - G4;
        #pragma unroll
        for (int k = 0; k < HID; ++k) wrow[k] = w_hm[j * HID + k];
        bhm = b_hm[j];
    }
    float c = 0.0f;                                  // cell state, threads 256..319
    __syncthreads();

    // ---- stage D: the sequential recurrence ----
    for (int t = 0; t < L; ++t) {
        const int tok = (int)sh_seq[t];

        if (tid >= G4) {                             // waves 8-9: hm & mx
            const int j = tid - G4;
            float a0 = bhm, a1 = 0.0f, a2 = 0.0f, a3 = 0.0f;
            #pragma unroll
            for (int k = 0; k < HID; k += 4) {
                a0 = __builtin_fmaf(wrow[k + 0], sh_h[k + 0], a0);
                a1 = __builtin_fmaf(wrow[k + 1], sh_h[k + 1], a1);
                a2 = __builtin_fmaf(wrow[k + 2], sh_h[k + 2], a2);
                a3 = __builtin_fmaf(wrow[k + 3], sh_h[k + 3], a3);
            }
            sh_mx[j] = sh_xim[tok][j] * ((a0 + a1) + (a2 + a3));
        }
        __syncthreads();

        if (tid < G4) {                              // waves 0-7: one gate each
            float a0 = sh_xihb[tok][tid], a1 = 0.0f, a2 = 0.0f, a3 = 0.0f;
            #pragma unroll
            for (int k = 0; k < HID; k += 4) {
                a0 = __builtin_fmaf(wrow[k + 0], sh_mx[k + 0], a0);
                a1 = __builtin_fmaf(wrow[k + 1], sh_mx[k + 1], a1);
                a2 = __builtin_fmaf(wrow[k + 2], sh_mx[k + 2], a2);
                a3 = __builtin_fmaf(wrow[k + 3], sh_mx[k + 3], a3);
            }
            sh_gates[tid] = (a0 + a1) + (a2 + a3);
        }
        __syncthreads();

        if (tid >= G4) {                             // waves 8-9: state update
            const int j = tid - G4;
            const float i_ = fast_sigmoid(sh_gates[j]);
            const float f_ = fast_sigmoid(sh_gates[j + HID]);
            const float g_ = fast_tanh   (sh_gates[j + 2 * HID]);
            const float o_ = fast_sigmoid(sh_gates[j + 3 * HID]);
            c = __builtin_fmaf(f_, c, i_ * g_);
            sh_h[j] = o_ * fast_tanh(c);
        }
        __syncthreads();
    }

    // ---- stage E: classifier head on c_last ----
    if (tid >= G4) sh_c[tid - G4] = c;
    __syncthreads();
    if (tid < 8) {                                   // OUT = 8
        float a0 = b_out[tid], a1 = 0.0f, a2 = 0.0f, a3 = 0.0f;
        #pragma unroll
        for (int k = 0; k < HID; k += 4) {
            a0 = __builtin_fmaf(W_out[tid * HID + k + 0], sh_c[k + 0], a0);
            a1 = __builtin_fmaf(W_out[tid * HID + k + 1], sh_c[k + 1], a1);
            a2 = __builtin_fmaf(W_out[tid * HID + k + 2], sh_c[k + 2], a2);
            a3 = __builtin_fmaf(W_out[tid * HID + k + 3], sh_c[k + 3], a3);
        }
        out[tid] = (a0 + a1) + (a2 + a3);
    }
}

extern "C" void kernel_launch(void* const* d_in, const int* in_sizes, int n_in,
                              void* d_out, int out_size, void* d_ws, size_t ws_size,
                              hipStream_t stream) {
    (void)n_in; (void)out_size; (void)d_ws; (void)ws_size;
    const int*   seq   = (const int*)  d_in[0];
    const float* emb   = (const float*)d_in[1];
    const float* w_im  = (const float*)d_in[2];
    const float* b_im  = (const float*)d_in[3];
    const float* w_hm  = (const float*)d_in[4];
    const float* b_hm  = (const float*)d_in[5];
    const float* W_ih  = (const float*)d_in[6];
    const float* b_ih  = (const float*)d_in[7];
    const float* W_hh  = (const float*)d_in[8];
    const float* b_hh  = (const float*)d_in[9];
    const float* W_out = (const float*)d_in[10];
    const float* b_out = (const float*)d_in[11];

    int L = in_sizes[0];
    if (L > LMAX) L = LMAX;

    hipLaunchKernelGGL(mlstm_seq_kernel, dim3(1), dim3(NTHR), 0, stream,
                       seq, emb, w_im, b_im, w_hm, b_hm,
                       W_ih, b_ih, W_hh, b_hh, W_out, b_out,
                       (float*)d_out, L);
}